// NotEnoughSleepAIModel_90735479095437
// MI455X (gfx1250) — compile-verified
//
#include <hip/hip_runtime.h>
#include <hip/hip_bf16.h>

#define NPIX   (1u << 20)   // 1024*1024 spatial positions per head
#define WBITS  10u          // W = 1024
#define WMASK  1023u
#define TPB    256u
#define HALF_PI_F 1.5707963267948966f

typedef __attribute__((address_space(3))) float lds_float_t;
typedef __attribute__((ext_vector_type(4))) float v4f;   // native vector: OK for nontemporal builtins

__device__ __forceinline__ float fast_sigmoid(float x) {
    return 1.0f / (1.0f + __expf(-x));
}

// Async LDS -> global 128-bit store (gfx1250 ASYNCcnt path), non-temporal:
// streamed output, no reuse -> don't let it churn WGP$/L2.
// vaddr = 64-bit global address (VGPR pair), vsrc = raw LDS byte offset (VGPR).
__device__ __forceinline__ void async_store_lds_b128_nt(const float* gaddr, unsigned lds_off) {
    asm volatile("global_store_async_from_lds_b128 %0, %1, off th:TH_STORE_NT"
                 :: "v"(gaddr), "v"(lds_off)
                 : "memory");
}

__device__ __forceinline__ float ld_nt(const float* p) {
    return __builtin_nontemporal_load(p);
}

__global__ __launch_bounds__(TPB) void decode_heads_kernel(
    const float* __restrict__ obj1_in,  const float* __restrict__ cls1_in,
    const float* __restrict__ bb1_in,
    const float* __restrict__ obj2_in,  const float* __restrict__ cls2_in,
    const float* __restrict__ bb2_in,
    const float* __restrict__ anchor_orients,   // (2)
    const float* __restrict__ anchor_dims,      // (3)
    float* __restrict__ boxes1_o,  float* __restrict__ scores1_o,
    float* __restrict__ obj1_o,    float* __restrict__ mask1_o,
    float* __restrict__ boxes2_o,  float* __restrict__ scores2_o,
    float* __restrict__ obj2_o,    float* __restrict__ mask2_o)
{
    __shared__ __align__(16) float sbox[TPB * 7u];   // 7168 B staging tile

    const unsigned tid  = threadIdx.x;
    const unsigned head = blockIdx.y;                // uniform -> s_cselect
    const unsigned i    = blockIdx.x * TPB + tid;

    const float* __restrict__ obj_in = head ? obj2_in : obj1_in;
    const float* __restrict__ cls_in = head ? cls2_in : cls1_in;
    const float* __restrict__ bb_in  = head ? bb2_in  : bb1_in;
    float* __restrict__ boxes_out  = head ? boxes2_o  : boxes1_o;
    float* __restrict__ scores_out = head ? scores2_o : scores1_o;
    float* __restrict__ obj_out    = head ? obj2_o    : obj1_o;
    float* __restrict__ mask_out   = head ? mask2_o   : mask1_o;

    // Uniform scalars (s_load, broadcast)
    const float orient = anchor_orients[head];
    const float ad0 = anchor_dims[0];
    const float ad1 = anchor_dims[1];
    const float ad2 = anchor_dims[2];

    // Objectness + mask (streamed once -> NT load)
    const float o = ld_nt(&obj_in[i]);
    const float m = (o >= 0.9f) ? 1.0f : 0.0f;

    // Planar bbox reads: each channel plane is fully coalesced across the wave.
    float b[7];
    #pragma unroll
    for (int c = 0; c < 7; ++c) b[c] = ld_nt(&bb_in[(size_t)c * NPIX + i]);

    // Hint the streamer: next block's bbox plane (gfx1250 global_prefetch_b8).
    __builtin_prefetch(&bb_in[i + TPB], 0, 0);

    // grid = (row, col) + 0.5
    const float fi = (float)(i >> WBITS) + 0.5f;
    const float fj = (float)(i & WMASK) + 0.5f;

    const float bx0 = (orient + fast_sigmoid(b[0]) * HALF_PI_F) * m;
    const float bx1 = (fast_sigmoid(b[1]) + fi) * m;
    const float bx2 = (fast_sigmoid(b[2]) + fj) * m;
    const float bx3 = fast_sigmoid(b[3]) * m;
    const float bx4 = (ad0 * __expf(b[4])) * m;
    const float bx5 = (ad1 * __expf(b[5])) * m;
    const float bx6 = (ad2 * __expf(b[6])) * m;

    // Stage the transposed (row-major) box rows in LDS.
    float* srow = &sbox[tid * 7u];
    srow[0] = bx0; srow[1] = bx1; srow[2] = bx2; srow[3] = bx3;
    srow[4] = bx4; srow[5] = bx5; srow[6] = bx6;

    // Scores: rows are 32 B and 32 B-aligned -> two dense NT 128-bit stores.
    {
        v4f s0, s1;
        s0.x = ld_nt(&cls_in[0u * NPIX + i]) * m;
        s0.y = ld_nt(&cls_in[1u * NPIX + i]) * m;
        s0.z = ld_nt(&cls_in[2u * NPIX + i]) * m;
        s0.w = ld_nt(&cls_in[3u * NPIX + i]) * m;
        s1.x = ld_nt(&cls_in[4u * NPIX + i]) * m;
        s1.y = ld_nt(&cls_in[5u * NPIX + i]) * m;
        s1.z = ld_nt(&cls_in[6u * NPIX + i]) * m;
        s1.w = ld_nt(&cls_in[7u * NPIX + i]) * m;
        v4f* so = (v4f*)(scores_out + (size_t)i * 8u);
        __builtin_nontemporal_store(s0, so + 0);
        __builtin_nontemporal_store(s1, so + 1);
    }

    __builtin_nontemporal_store(o * m, &obj_out[i]);
    __builtin_nontemporal_store(m,     &mask_out[i]);

    __syncthreads();   // all ds_stores visible before async drain

    // Drain the contiguous 7168 B staged block with async LDS->global b128.
    // 448 16-byte chunks, 256 lanes: every lane does chunk tid, lanes < 192
    // also do chunk 256+tid. Block base = blk*7168 B (16 B aligned).
    {
        const float* gbase = boxes_out + (size_t)blockIdx.x * (TPB * 7u);

        unsigned k0 = tid;
        unsigned l0 = (unsigned)(size_t)(lds_float_t*)&sbox[k0 * 4u];
        async_store_lds_b128_nt(gbase + (size_t)k0 * 4u, l0);

        if (tid < 192u) {
            unsigned k1 = TPB + tid;
            unsigned l1 = (unsigned)(size_t)(lds_float_t*)&sbox[k1 * 4u];
            async_store_lds_b128_nt(gbase + (size_t)k1 * 4u, l1);
        }
    }

    // Ensure async transfers out of LDS complete before LDS is deallocated.
    asm volatile("s_wait_asynccnt 0" ::: "memory");
}

extern "C" void kernel_launch(void* const* d_in, const int* in_sizes, int n_in,
                              void* d_out, int out_size, void* d_ws, size_t ws_size,
                              hipStream_t stream) {
    (void)in_sizes; (void)n_in; (void)out_size; (void)d_ws; (void)ws_size;

    const float* obj1 = (const float*)d_in[0];
    const float* cls1 = (const float*)d_in[1];
    const float* bb1  = (const float*)d_in[2];
    const float* obj2 = (const float*)d_in[3];
    const float* cls2 = (const float*)d_in[4];
    const float* bb2  = (const float*)d_in[5];
    const float* anchor_orients = (const float*)d_in[6];
    const float* anchor_dims    = (const float*)d_in[7];

    float* out = (float*)d_out;
    const size_t N = (size_t)NPIX;

    // Output layout (flat, return order):
    // boxes1[7N] scores1[8N] obj1[N] mask1[N] boxes2[7N] scores2[8N] obj2[N] mask2[N]
    float* boxes1_o  = out;
    float* scores1_o = out + 7u * N;
    float* obj1_o    = out + 15u * N;
    float* mask1_o   = out + 16u * N;
    float* boxes2_o  = out + 17u * N;
    float* scores2_o = out + 24u * N;
    float* obj2_o    = out + 32u * N;
    float* mask2_o   = out + 33u * N;

    const dim3 grid(NPIX / TPB, 2);   // 4096 blocks per head, y = head
    const dim3 block(TPB);            // 8 wave32s per block

    decode_heads_kernel<<<grid, block, 0, stream>>>(
        obj1, cls1, bb1, obj2, cls2, bb2,
        anchor_orients, anchor_dims,
        boxes1_o, scores1_o, obj1_o, mask1_o,
        boxes2_o, scores2_o, obj2_o, mask2_o);
}